// RootLayer_77120432767513
// MI455X (gfx1250) — compile-verified
//
#include <hip/hip_runtime.h>
#include <math.h>

typedef __attribute__((ext_vector_type(2))) float v2f;
typedef __attribute__((ext_vector_type(8))) float v8f;

#define FDIM 4096   // P*N
#define BDIM 2048
#define CDIM 10
#define CPAD 16
#define KSEG 16                 // total K segments
#define WAVES_PER_BLOCK 8
#define KSEG_Y (KSEG / WAVES_PER_BLOCK)
#define KLEN (FDIM / KSEG)      // 256 K-values per wave

// ---------------- kernel 1: row softmax of weight into padded 16 x F buffer ----
__global__ __launch_bounds__(256) void softmax_rows(const float* __restrict__ w,
                                                    float* __restrict__ smw) {
    const int r = blockIdx.x;          // 0..15 (rows 10..15 are zero padding)
    const int tid = threadIdx.x;
    __shared__ float red[256];
    if (r >= CDIM) {
        for (int f = tid; f < FDIM; f += 256) smw[r * FDIM + f] = 0.0f;
        return;
    }
    const float* wr = w + r * FDIM;
    float m = -3.4e38f;
    for (int f = tid; f < FDIM; f += 256) m = fmaxf(m, wr[f]);
    red[tid] = m; __syncthreads();
    for (int s = 128; s > 0; s >>= 1) {
        if (tid < s) red[tid] = fmaxf(red[tid], red[tid + s]);
        __syncthreads();
    }
    m = red[0]; __syncthreads();
    float sum = 0.f;
    for (int f = tid; f < FDIM; f += 256) sum += __expf(wr[f] - m);
    red[tid] = sum; __syncthreads();
    for (int s = 128; s > 0; s >>= 1) {
        if (tid < s) red[tid] += red[tid + s];
        __syncthreads();
    }
    const float inv = 1.0f / red[0];
    for (int f = tid; f < FDIM; f += 256) smw[r * FDIM + f] = __expf(wr[f] - m) * inv;
}

// ---------------- kernel 2: partial[kseg] = exp(x) @ softmax(W)^T via f32 WMMA --
__global__ __launch_bounds__(256) void lse_gemm(const float* __restrict__ x,
                                                const float* __restrict__ smw,
                                                float* __restrict__ partial) {
    const int lane  = threadIdx.x & 31;
    const int wave  = threadIdx.x >> 5;
    const int l15   = lane & 15;
    const int khalf = lane >> 4;                 // 0: K={0,1}, 1: K={2,3}
    const int mbase = blockIdx.x * 16;
    const int kseg  = blockIdx.y * WAVES_PER_BLOCK + wave;
    const int k0    = kseg * KLEN;

    // A fragment source: row (mbase + l15), f32 pair at k + khalf*2
    const float* xp = x   + (size_t)(mbase + l15) * FDIM + k0 + khalf * 2;
    // B fragment source: B[k][n] = smw[n][k]  (GEMM with W^T)
    const float* bp = smw + (size_t)l15 * FDIM + k0 + khalf * 2;

    v8f acc0 = {0.f, 0.f, 0.f, 0.f, 0.f, 0.f, 0.f, 0.f};
    v8f acc1 = {0.f, 0.f, 0.f, 0.f, 0.f, 0.f, 0.f, 0.f};

    // Constant trip count -> uniform scalar-counted loop, unrolls cleanly.
    // Two independent WMMA accumulation chains per iteration (K and K+4).
#pragma unroll 4
    for (int i = 0; i < KLEN / 8; ++i) {
        const float a00 = xp[0], a01 = xp[1];
        const float a10 = xp[4], a11 = xp[5];
        const float b00 = bp[0], b01 = bp[1];
        const float b10 = bp[4], b11 = bp[5];

        v2f A0; A0.x = __expf(a00); A0.y = __expf(a01);
        v2f A1; A1.x = __expf(a10); A1.y = __expf(a11);
        v2f B0; B0.x = b00; B0.y = b01;
        v2f B1; B1.x = b10; B1.y = b11;

        acc0 = __builtin_amdgcn_wmma_f32_16x16x4_f32(
            false, A0, false, B0, (short)0, acc0, false, false);
        acc1 = __builtin_amdgcn_wmma_f32_16x16x4_f32(
            false, A1, false, B1, (short)0, acc1, false, false);

        xp += 8;
        bp += 8;
    }

    const v8f acc = acc0 + acc1;   // exact: disjoint K partial sums

    // C/D layout: VGPR r -> M = r + khalf*8, N = l15
    float* p = partial + (size_t)kseg * (BDIM * CPAD);
#pragma unroll
    for (int r = 0; r < 8; ++r) {
        const int row = mbase + r + khalf * 8;
        p[(size_t)row * CPAD + l15] = acc[r];
    }
}

// ---------------- kernel 3: sum partials over K segments, take log -------------
__global__ __launch_bounds__(256) void final_log(const float* __restrict__ partial,
                                                 float* __restrict__ out) {
    const int idx = blockIdx.x * 256 + threadIdx.x;
    if (idx >= BDIM * CDIM) return;
    const int b = idx / CDIM;
    const int c = idx % CDIM;
    float s = 0.f;
#pragma unroll
    for (int g = 0; g < KSEG; ++g)
        s += partial[(size_t)g * (BDIM * CPAD) + (size_t)b * CPAD + c];
    out[idx] = logf(s);
}

extern "C" void kernel_launch(void* const* d_in, const int* in_sizes, int n_in,
                              void* d_out, int out_size, void* d_ws, size_t ws_size,
                              hipStream_t stream) {
    const float* x = (const float*)d_in[0];   // (2048, 64, 64) f32
    const float* w = (const float*)d_in[1];   // (10, 4096)     f32
    float* out = (float*)d_out;               // (2048, 10)     f32

    float* smw     = (float*)d_ws;               // 16 * 4096 floats (256 KB)
    float* partial = smw + (size_t)CPAD * FDIM;  // KSEG * 2048 * 16 floats (2 MB)

    softmax_rows<<<CPAD, 256, 0, stream>>>(w, smw);

    dim3 grid(BDIM / 16, KSEG_Y);
    lse_gemm<<<grid, 256, 0, stream>>>(x, smw, partial);

    final_log<<<(BDIM * CDIM + 255) / 256, 256, 0, stream>>>(partial, out);
}